// HMM_33818572489376
// MI455X (gfx1250) — compile-verified
//
#include <hip/hip_runtime.h>
#include <math.h>

// ---------------- CDNA5 WMMA f32 16x16x4 tile helpers (wave32) ----------------
typedef float v2f __attribute__((ext_vector_type(2)));
typedef float v8f __attribute__((ext_vector_type(8)));

#define TSTEPS   4096
#define OD       64      // observation dim
#define DVALID   66      // true augmented state dim
#define DP       80      // padded state dim (5 tiles of 16)
#define LDP      80      // P leading dim
#define LSL      65      // S/L leading dim (64x64 stored with pad)
#define LDX      96      // CP/X leading dim (6 tiles; col 80 carries residual)
#define RCOL     80
#define BLOCK    256
#define NWAVE    8
#define LOG2PI_F 1.8378770664093453f

// C += A(16xK) * B(Kx16), elements supplied by functors (negate in functor to
// subtract).  Layout per ISA 7.12.2: lane = 16*half + mn ;
// A frag: a[v]=A[mn][k0+2*half+v], B frag: b[v]=B[k0+2*half+v][mn],
// C/D frag: c[v]=C[v+8*half][mn].
template <typename FA, typename FB>
__device__ inline v8f tile_mac(v8f c, int K, FA lda, FB ldb) {
  const int lane = threadIdx.x & 31;
  const int mn   = lane & 15;
  const int half = lane >> 4;
#pragma unroll
  for (int k0 = 0; k0 < 16; k0 += 4) {   // K==16 fast path handled by caller loop
    (void)k0;
  }
  for (int k0 = 0; k0 < K; k0 += 4) {
    const int ka = k0 + 2 * half;
    v2f a, b;
    a[0] = lda(mn, ka);     a[1] = lda(mn, ka + 1);
    b[0] = ldb(ka, mn);     b[1] = ldb(ka + 1, mn);
    c = __builtin_amdgcn_wmma_f32_16x16x4_f32(false, a, false, b,
                                              (short)0, c, false, false);
  }
  return c;
}

__device__ inline v8f tile_zero() {
  v8f c;
#pragma unroll
  for (int v = 0; v < 8; ++v) c[v] = 0.0f;
  return c;
}

__device__ inline v8f tile_load(const float* base, int ld) {
  const int lane = threadIdx.x & 31;
  const int n = lane & 15, h = lane >> 4;
  v8f c;
#pragma unroll
  for (int v = 0; v < 8; ++v) c[v] = base[(v + 8 * h) * ld + n];
  return c;
}

__device__ inline void tile_store(float* base, int ld, v8f c) {
  const int lane = threadIdx.x & 31;
  const int n = lane & 15, h = lane >> 4;
#pragma unroll
  for (int v = 0; v < 8; ++v) base[(v + 8 * h) * ld + n] = c[v];
}

__device__ inline float wave_sum(float v) {
  v += __shfl_down(v, 16, 32);
  v += __shfl_down(v,  8, 32);
  v += __shfl_down(v,  4, 32);
  v += __shfl_down(v,  2, 32);
  v += __shfl_down(v,  1, 32);
  return v;
}

// ------------------------------ persistent Kalman kernel ------------------------------
__global__ void __launch_bounds__(BLOCK)
kalman_hmm_ll(const float* __restrict__ track,
              const float* __restrict__ bias_scales,
              const float* __restrict__ obs_noise,
              const float* __restrict__ trans_noise,
              const float* __restrict__ transition_param,
              float* __restrict__ out) {
  extern __shared__ float smem[];
  float* P    = smem;                    // DP*LDP
  float* CP   = P    + DP * LDP;         // OD*LDX  (C@P, col 80 = residual r)
  float* X    = CP   + OD * LDX;         // OD*LDX  (S^-1 @ CP ; col 80 = alpha)
  float* SL   = X    + OD * LDX;         // OD*LSL  (innovation cov -> Cholesky L)
  float* MI   = SL   + OD * LSL;         // 64*16   (L11^-1 for the 4 diag blocks)
  float* mvec = MI   + OD * 16;          // DP
  float* rres = mvec + DP;               // OD
  float* tmpr = rres + OD;               // 2*DP
  float* scal = tmpr + 2 * DP;           // 4

  const int tid  = threadIdx.x;
  const int wave = __builtin_amdgcn_readfirstlane(tid >> 5);  // scalar wave id
  const int lane = tid & 31;

  const float p0 = transition_param[0];
  const float p1 = transition_param[1];
  const float q  = trans_noise[0] * trans_noise[0];
  const float r2 = obs_noise[0]   * obs_noise[0];

  // ---- init: m=0, P = diag(1,1, tile(bias_scales,32)), ll=0 ----
  for (int i = tid; i < DP * LDP; i += BLOCK) P[i] = 0.0f;
  for (int i = tid; i < DP; i += BLOCK)       mvec[i] = 0.0f;
  if (tid == 0) scal[0] = 0.0f;
  __syncthreads();
  if (tid < DVALID) {
    float d = (tid < 2) ? 1.0f : bias_scales[(tid - 2) & 1];
    P[tid * LDP + tid] = d;
  }
  __syncthreads();

  for (int t = 0; t < TSTEPS; ++t) {
    // ================= predict: m=F m ; P = F P F^T + Q =================
    if (tid < DP) {
      tmpr[tid]      = p0 * P[1 * LDP + tid];
      tmpr[DP + tid] = p1 * P[0 * LDP + tid];
    }
    __syncthreads();
    if (tid < DP) { P[0 * LDP + tid] = tmpr[tid]; P[1 * LDP + tid] = tmpr[DP + tid]; }
    __syncthreads();
    if (tid < DP) {
      float c0 = p0 * P[tid * LDP + 1];
      float c1 = p1 * P[tid * LDP + 0];
      P[tid * LDP + 0] = c0;
      P[tid * LDP + 1] = c1;
    }
    __syncthreads();
    if (tid == 0) {
      P[0] += q; P[LDP + 1] += q;
      float nm0 = p0 * mvec[1], nm1 = p1 * mvec[0];
      mvec[0] = nm0; mvec[1] = nm1;
    }
    __syncthreads();

    // ================= residual r = y - C m =================
    if (tid < OD)
      rres[tid] = track[(size_t)t * OD + tid] - (mvec[tid >> 5] + mvec[2 + tid]);
    __syncthreads();

    // ========== CP = C@P (+ residual as column 80); X = copy of CP ==========
    for (int idx = tid; idx < OD * LDX; idx += BLOCK) {
      int i = idx / LDX, c = idx - i * LDX;
      float v;
      if      (c < DP)    v = P[(i >> 5) * LDP + c] + P[(2 + i) * LDP + c];
      else if (c == RCOL) v = rres[i];
      else                v = 0.0f;
      CP[idx] = v;
      X[idx]  = v;
    }
    __syncthreads();

    // ========== S = CP @ C^T + r2*I ==========
    for (int idx = tid; idx < OD * OD; idx += BLOCK) {
      int i = idx >> 6, j = idx & 63;
      float v = CP[i * LDX + (j >> 5)] + CP[i * LDX + 2 + j];
      if (i == j) v += r2;
      SL[i * LSL + j] = v;
    }
    __syncthreads();

    // ========== blocked Cholesky of S (4 panels of 16) ==========
#pragma unroll
    for (int kb = 0; kb < 4; ++kb) {
      const int base = kb * 16;
      // (a) 16x16 diagonal factor: register-resident in wave 0, shfl broadcasts
      if (wave == 0 && lane < 16) {
        float a[16];
        const int row = base + lane;
#pragma unroll
        for (int j = 0; j < 16; ++j) a[j] = SL[row * LSL + base + j];
#pragma unroll
        for (int j = 0; j < 16; ++j) {
          float dj = sqrtf(__shfl(a[j], j, 32));
          a[j] = (lane == j) ? dj : (a[j] / dj);
#pragma unroll
          for (int k = j + 1; k < 16; ++k) {
            float lkj = __shfl(a[j], k, 32);
            a[k] -= a[j] * lkj;
          }
        }
#pragma unroll
        for (int j = 0; j < 16; ++j)
          if (j <= lane) SL[row * LSL + base + j] = a[j];
      }
      __syncthreads();
      // (a2) Minv = L11^-1 : 16 threads, one per column (L11^-1 is lower-tri)
      if (tid < 16) {
        const int c = tid;
        float x[16];
#pragma unroll
        for (int j = 0; j < 16; ++j) {
          float acc = (j == c) ? 1.0f : 0.0f;
#pragma unroll
          for (int k = 0; k < j; ++k) acc -= SL[(base + j) * LSL + base + k] * x[k];
          x[j] = acc / SL[(base + j) * LSL + base + j];
        }
#pragma unroll
        for (int j = 0; j < 16; ++j) MI[(base + j) * 16 + c] = x[j];
      }
      __syncthreads();
      if (kb < 3) {
        // (b) panel TRSM via WMMA: L21 = A21 * Minv^T (in-place per tile)
#pragma unroll
        for (int ib = kb + 1; ib < 4; ++ib) {
          if (((ib - kb - 1) % NWAVE) == wave) {
            float* cb = &SL[(ib * 16) * LSL + base];
            v8f c = tile_zero();
            c = tile_mac(c, 16,
                  [&](int mm, int kk) { return SL[(ib * 16 + mm) * LSL + base + kk]; },
                  [&](int kk, int nn) { return MI[(base + nn) * 16 + kk]; });
            tile_store(cb, LSL, c);
          }
        }
        __syncthreads();
        // (c) trailing SYRK: A22 -= L21 L21^T  via WMMA
        {
          int cnt = 0;
#pragma unroll
          for (int ib = kb + 1; ib < 4; ++ib)
#pragma unroll
            for (int jb = kb + 1; jb <= ib; ++jb) {
              if ((cnt++ % NWAVE) == wave) {
                float* cb = &SL[(ib * 16) * LSL + jb * 16];
                const float* Ab = &SL[(ib * 16) * LSL + base];
                const float* Bb = &SL[(jb * 16) * LSL + base];
                v8f c = tile_load(cb, LSL);
                c = tile_mac(c, 16,
                      [&](int mm, int kk) { return -Ab[mm * LSL + kk]; },
                      [&](int kk, int nn) { return  Bb[nn * LSL + kk]; });
                tile_store(cb, LSL, c);
              }
            }
        }
        __syncthreads();
      }
    }

    // ========== forward solve: X <- L^-1 X  (96 RHS columns, all WMMA) ==========
#pragma unroll
    for (int ib = 0; ib < 4; ++ib) {
      const int base = ib * 16;
      // diag: X_blk = Minv @ X_blk  (in-place per tile: loads precede stores)
#pragma unroll
      for (int nt = 0; nt < 6; ++nt) {
        if ((nt % NWAVE) == wave) {
          float* cb = &X[base * LDX + nt * 16];
          v8f c = tile_zero();
          c = tile_mac(c, 16,
                [&](int mm, int kk) { return MI[(base + mm) * 16 + kk]; },
                [&](int kk, int nn) { return X[(base + kk) * LDX + nt * 16 + nn]; });
          tile_store(cb, LDX, c);
        }
      }
      __syncthreads();
      if (ib < 3) {
        int cnt = 0;
#pragma unroll
        for (int jb = ib + 1; jb < 4; ++jb)
#pragma unroll
          for (int nt = 0; nt < 6; ++nt) {
            if ((cnt++ % NWAVE) == wave) {
              float* cb = &X[(jb * 16) * LDX + nt * 16];
              const float* Ab = &SL[(jb * 16) * LSL + base];
              const float* Bb = &X[base * LDX + nt * 16];
              v8f c = tile_load(cb, LDX);
              c = tile_mac(c, 16,
                    [&](int mm, int kk) { return -Ab[mm * LSL + kk]; },
                    [&](int kk, int nn) { return  Bb[kk * LDX + nn]; });
              tile_store(cb, LDX, c);
            }
          }
        __syncthreads();
      }
    }

    // ========== backward solve: X <- L^-T X  (all WMMA) ==========
#pragma unroll
    for (int ib = 3; ib >= 0; --ib) {
      const int base = ib * 16;
      // diag: X_blk = Minv^T @ X_blk
#pragma unroll
      for (int nt = 0; nt < 6; ++nt) {
        if ((nt % NWAVE) == wave) {
          float* cb = &X[base * LDX + nt * 16];
          v8f c = tile_zero();
          c = tile_mac(c, 16,
                [&](int mm, int kk) { return MI[(base + kk) * 16 + mm]; },
                [&](int kk, int nn) { return X[(base + kk) * LDX + nt * 16 + nn]; });
          tile_store(cb, LDX, c);
        }
      }
      __syncthreads();
      if (ib > 0) {
        int cnt = 0;
#pragma unroll
        for (int jb = 0; jb < ib; ++jb)
#pragma unroll
          for (int nt = 0; nt < 6; ++nt) {
            if ((cnt++ % NWAVE) == wave) {
              float* cb = &X[(jb * 16) * LDX + nt * 16];
              const float* Ab = &SL[(ib * 16) * LSL + jb * 16];  // transposed panel
              const float* Bb = &X[(ib * 16) * LDX + nt * 16];
              v8f c = tile_load(cb, LDX);
              c = tile_mac(c, 16,
                    [&](int mm, int kk) { return -Ab[kk * LSL + mm]; },
                    [&](int kk, int nn) { return  Bb[kk * LDX + nn]; });
              tile_store(cb, LDX, c);
            }
          }
        __syncthreads();
      }
    }

    // ========== log-likelihood: ll += -0.5 r.alpha - sum log diag L - c ==========
    if (wave == 0) {
      float dp = rres[lane] * X[lane * LDX + RCOL]
               + rres[lane + 32] * X[(lane + 32) * LDX + RCOL];
      float sl = logf(SL[lane * LSL + lane])
               + logf(SL[(lane + 32) * LSL + lane + 32]);
      dp = wave_sum(dp);
      sl = wave_sum(sl);
      if (lane == 0)
        scal[0] += -0.5f * dp - sl - 0.5f * (float)OD * LOG2PI_F;
    }
    __syncthreads();

    // ========== mean update: m += X^T r  (K = X^T) ==========
    if (tid < DP) {
      float acc = mvec[tid];
      for (int k = 0; k < OD; ++k) acc += X[k * LDX + tid] * rres[k];
      mvec[tid] = acc;
    }
    __syncthreads();

    // ========== covariance downdate: P -= X^T @ CP  (WMMA, 25 tiles, K=64) ==========
    {
      int cnt = 0;
#pragma unroll
      for (int mt = 0; mt < 5; ++mt)
#pragma unroll
        for (int nt = 0; nt < 5; ++nt) {
          if ((cnt++ % NWAVE) == wave) {
            float* cb = &P[(mt * 16) * LDP + nt * 16];
            v8f c = tile_load(cb, LDP);
            c = tile_mac(c, OD,
                  [&](int mm, int kk) { return -X[kk * LDX + mt * 16 + mm]; },
                  [&](int kk, int nn) { return  CP[kk * LDX + nt * 16 + nn]; });
            tile_store(cb, LDP, c);
          }
        }
    }
    __syncthreads();
    // symmetrize P
    for (int idx = tid; idx < DP * LDP; idx += BLOCK) {
      int i = idx / LDP, j = idx - i * LDP;
      if (i < j) {
        float v = 0.5f * (P[i * LDP + j] + P[j * LDP + i]);
        P[i * LDP + j] = v;
        P[j * LDP + i] = v;
      }
    }
    __syncthreads();
  }

  if (tid == 0) out[0] = scal[0];
}

extern "C" void kernel_launch(void* const* d_in, const int* in_sizes, int n_in,
                              void* d_out, int out_size, void* d_ws, size_t ws_size,
                              hipStream_t stream) {
  (void)in_sizes; (void)n_in; (void)out_size; (void)d_ws; (void)ws_size;
  const float* track            = (const float*)d_in[0];
  const float* bias_scales      = (const float*)d_in[1];
  const float* obs_noise        = (const float*)d_in[2];
  const float* trans_noise      = (const float*)d_in[3];
  const float* transition_param = (const float*)d_in[4];
  float* out = (float*)d_out;

  const size_t smem =
      (size_t)(DP * LDP + 2 * OD * LDX + OD * LSL + OD * 16 + DP + OD + 2 * DP + 4) *
      sizeof(float);  // ~96.7 KB of the 320 KB WGP LDS

  kalman_hmm_ll<<<dim3(1), dim3(BLOCK), smem, stream>>>(
      track, bias_scales, obs_noise, trans_noise, transition_param, out);
}